// CapsuleNet_50440095924518
// MI455X (gfx1250) — compile-verified
//
#include <hip/hip_runtime.h>
#include <math.h>

// ---------------- problem constants ----------------
#define BATCH 128
#define ACH   128          // conv1 out channels
#define H1    14           // conv1 out spatial
#define HW1   196          // 14*14
#define PCCH  544          // B*D + B = 32*16 + 32
#define BCAP  32           // B capsule types
#define DDIM  16           // D
#define KOUT  6            // K
#define SS    36           // K*K
#define CC    10           // C classes
#define NNN   1152         // B*K*K
#define KGEMM 1152         // prim conv reduction = 128*3*3
#define GN    4608         // GEMM N = BATCH*SS
#define LN2   0.6931471805599453f
#define LN2PI 1.8378770664093453f
#define BN_EPS 1e-5f
#define EPSF   1e-8f

typedef __attribute__((ext_vector_type(2))) float v2f;
typedef __attribute__((ext_vector_type(8))) float v8f;

// ---------------- digamma (args >= ~1.0 in this model) ----------------
__device__ __forceinline__ float digammaf(float x) {
  float r = 0.f;
  while (x < 6.f) { r -= 1.f / x; x += 1.f; }
  float inv = 1.f / x, inv2 = inv * inv;
  return r + logf(x) - 0.5f * inv
           - inv2 * (1.f/12.f - inv2 * (1.f/120.f - inv2 * (1.f/252.f)));
}

// ---------------- stage 1: conv1 (5x5 s2 VALID) ----------------
__global__ void capsnet_conv1(const float* __restrict__ x,
                              const float* __restrict__ w,
                              float* __restrict__ h) {
  int gid = blockIdx.x * blockDim.x + threadIdx.x;
  if (gid >= BATCH * ACH * HW1) return;
  int p  = gid % HW1;
  int t  = gid / HW1;
  int oc = t % ACH;
  int bb = t / ACH;
  int oy = p / H1, ox = p % H1;
  const float* xb = x + (size_t)bb * 3 * 1024;
  const float* wc = w + (size_t)oc * 75;
  float acc = 0.f;
#pragma unroll
  for (int ci = 0; ci < 3; ci++)
#pragma unroll
    for (int ky = 0; ky < 5; ky++) {
      const float* xr = xb + ci * 1024 + (2 * oy + ky) * 32 + 2 * ox;
      const float* wr = wc + ci * 25 + ky * 5;
#pragma unroll
      for (int kx = 0; kx < 5; kx++) acc = fmaf(xr[kx], wr[kx], acc);
    }
  h[gid] = acc;
}

// ---------------- per-channel mean/invstd kernels ----------------
__global__ __launch_bounds__(256) void capsnet_stats_conv1(const float* __restrict__ h,
                                                           float* __restrict__ st) {
  __shared__ float r1[256], r2[256];
  int c = blockIdx.x;
  float s = 0.f, s2 = 0.f;
  for (int i = threadIdx.x; i < BATCH * HW1; i += 256) {
    int bb = i / HW1, p = i % HW1;
    float v = h[((size_t)bb * ACH + c) * HW1 + p];
    s += v; s2 += v * v;
  }
  r1[threadIdx.x] = s; r2[threadIdx.x] = s2; __syncthreads();
  for (int o = 128; o > 0; o >>= 1) {
    if (threadIdx.x < o) { r1[threadIdx.x] += r1[threadIdx.x + o]; r2[threadIdx.x] += r2[threadIdx.x + o]; }
    __syncthreads();
  }
  if (threadIdx.x == 0) {
    float cnt = (float)(BATCH * HW1);
    float mu = r1[0] / cnt, var = r2[0] / cnt - mu * mu;
    st[c] = mu; st[ACH + c] = rsqrtf(var + BN_EPS);
  }
}

__global__ __launch_bounds__(256) void capsnet_stats_a(const float* __restrict__ pc,
                                                       float* __restrict__ st) {
  __shared__ float r1[256], r2[256];
  int c = blockIdx.x;  // 0..31
  float s = 0.f, s2 = 0.f;
  for (int i = threadIdx.x; i < BATCH * SS; i += 256) {
    int bb = i / SS, sp = i % SS;
    float v = pc[((size_t)bb * PCCH + 512 + c) * SS + sp];
    s += v; s2 += v * v;
  }
  r1[threadIdx.x] = s; r2[threadIdx.x] = s2; __syncthreads();
  for (int o = 128; o > 0; o >>= 1) {
    if (threadIdx.x < o) { r1[threadIdx.x] += r1[threadIdx.x + o]; r2[threadIdx.x] += r2[threadIdx.x + o]; }
    __syncthreads();
  }
  if (threadIdx.x == 0) {
    float cnt = (float)(BATCH * SS);
    float mu = r1[0] / cnt, var = r2[0] / cnt - mu * mu;
    st[c] = mu; st[BCAP + c] = rsqrtf(var + BN_EPS);
  }
}

__global__ __launch_bounds__(256) void capsnet_stats_p(const float* __restrict__ pc,
                                                       float* __restrict__ st) {
  __shared__ float r1[256], r2[256];
  int c = blockIdx.x;  // capsule type 0..31, reduce over (batch, D, K, K)
  float s = 0.f, s2 = 0.f;
  for (int i = threadIdx.x; i < BATCH * DDIM * SS; i += 256) {
    int bb = i / (DDIM * SS);
    int rem = i % (DDIM * SS);
    int d = rem / SS, sp = rem % SS;
    float v = pc[((size_t)bb * PCCH + c * DDIM + d) * SS + sp];
    s += v; s2 += v * v;
  }
  r1[threadIdx.x] = s; r2[threadIdx.x] = s2; __syncthreads();
  for (int o = 128; o > 0; o >>= 1) {
    if (threadIdx.x < o) { r1[threadIdx.x] += r1[threadIdx.x + o]; r2[threadIdx.x] += r2[threadIdx.x + o]; }
    __syncthreads();
  }
  if (threadIdx.x == 0) {
    float cnt = (float)(BATCH * DDIM * SS);
    float mu = r1[0] / cnt, var = r2[0] / cnt - mu * mu;
    st[c] = mu; st[BCAP + c] = rsqrtf(var + BN_EPS);
  }
}

// ---------------- stage 2: fused conv1-BN/ReLU + im2col ----------------
// col[k][n], k = ci*9 + ky*3 + kx (K=1152), n = bb*36 + oy*6 + ox (N=4608)
__global__ void capsnet_bn_im2col(const float* __restrict__ h, const float* __restrict__ st,
                                  const float* __restrict__ g, const float* __restrict__ be,
                                  float* __restrict__ col) {
  int gid = blockIdx.x * blockDim.x + threadIdx.x;
  if (gid >= KGEMM * GN) return;
  int n = gid % GN;
  int k = gid / GN;
  int ci = k / 9;
  int rr = k - ci * 9;
  int ky = rr / 3, kx = rr - ky * 3;
  int bb = n / SS, sp = n % SS;
  int oy = sp / KOUT, ox = sp % KOUT;
  float raw = h[((size_t)bb * ACH + ci) * HW1 + (2 * oy + ky) * H1 + 2 * ox + kx];
  float v = (raw - st[ci]) * st[ACH + ci] * g[ci] + be[ci];
  col[gid] = v > 0.f ? v : 0.f;
}

// ---------------- stage 3: prim conv GEMM via V_WMMA_F32_16X16X4_F32 ----------------
// Out[m, n] = sum_k W[m,k] * col[k,n],  M=544 (34 tiles), N=4608 (288 tiles), K=1152
// A layout (16x4 f32): lanes 0-15 rows M=0..15 with K={k0,k0+1}; lanes 16-31 K={k0+2,k0+3}
// B layout (4x16 f32): lanes 0-15 cols N=0..15 with K={k0,k0+1}; lanes 16-31 K={k0+2,k0+3}
// C/D (16x16 f32): VGPR j = row M=j (lanes 0-15) / M=j+8 (lanes 16-31), col = lane&15
__global__ __launch_bounds__(32) void capsnet_primconv_wmma(const float* __restrict__ col,
                                                            const float* __restrict__ w,
                                                            float* __restrict__ pc) {
  const int tileN = blockIdx.x;      // 0..287
  const int tileM = blockIdx.y;      // 0..33
  const int lane  = threadIdx.x;     // 0..31
  const int half  = lane >> 4;
  const int l16   = lane & 15;

  // A: this lane's weight row, pre-offset by half's K pair
  const float* wr = w + (size_t)(tileM * 16 + l16) * KGEMM + 2 * half;
  // B: this lane's im2col column, pre-offset by half's K pair
  const float* cb = col + (size_t)(2 * half) * GN + tileN * 16 + l16;

  v8f acc = 0.f;
#pragma unroll 4
  for (int k0 = 0; k0 < KGEMM; k0 += 4) {
    v2f av; av.x = wr[k0];              av.y = wr[k0 + 1];
    v2f bv; bv.x = cb[(size_t)k0 * GN]; bv.y = cb[(size_t)(k0 + 1) * GN];
    acc = __builtin_amdgcn_wmma_f32_16x16x4_f32(
        /*neg_a=*/false, av, /*neg_b=*/false, bv,
        /*c_mod=*/(short)0, acc, /*reuse_a=*/false, /*reuse_b=*/false);
  }
  // scatter the 16x16 tile: VGPR j -> row M = tileM*16 + j + half*8, col = this lane's n
  const int n  = tileN * 16 + l16;
  const int bb = n / SS, sp = n % SS;
  const int oy = sp / KOUT, ox = sp % KOUT;
#pragma unroll
  for (int j = 0; j < 8; j++) {
    int mrow = tileM * 16 + j + half * 8;
    pc[(((size_t)bb * PCCH + mrow) * KOUT + oy) * KOUT + ox] = acc[j];
  }
}

// ---------------- stage 4: activation BN + sigmoid -> a_i (b, N) ----------------
__global__ void capsnet_apply_a(const float* __restrict__ pc, const float* __restrict__ st,
                                const float* __restrict__ g, const float* __restrict__ be,
                                float* __restrict__ ai) {
  int gid = blockIdx.x * blockDim.x + threadIdx.x;
  if (gid >= BATCH * NNN) return;
  int bb = gid / NNN;
  int rem = gid % NNN;        // = i*36 + s
  int i = rem / SS, sp = rem % SS;
  float v = pc[((size_t)bb * PCCH + 512 + i) * SS + sp];
  float y = (v - st[i]) * st[BCAP + i] * g[i] + be[i];
  ai[gid] = 1.f / (1.f + expf(-y));
}

// ---------------- stage 5: pose BN + vote transform V ----------------
// V[b, n=(i,x,y), c, d=(p,r)] = sum_q W_ij[i,c,x,y,p,q] * posesN[b,i,q,r,x,y]
__global__ void capsnet_votes(const float* __restrict__ pc, const float* __restrict__ st,
                              const float* __restrict__ g, const float* __restrict__ be,
                              const float* __restrict__ Wij, float* __restrict__ V) {
  int gid = blockIdx.x * blockDim.x + threadIdx.x;
  if (gid >= BATCH * BCAP * SS * CC) return;
  int c  = gid % CC;
  int t  = gid / CC;
  int sp = t % SS;
  int t2 = t / SS;
  int i  = t2 % BCAP;
  int bb = t2 / BCAP;

  float mu = st[i], iv = st[BCAP + i], ga = g[i], bt = be[i];
  float pn[16];
#pragma unroll
  for (int d = 0; d < 16; d++) {
    float raw = pc[((size_t)bb * PCCH + i * DDIM + d) * SS + sp];
    pn[d] = (raw - mu) * iv * ga + bt;   // pn[q*4+r]
  }
  const float* wb = Wij + (((size_t)(i * CC + c)) * SS + sp) * 16;  // wb[p*4+q]
  int n = i * SS + sp;
  float* vout = V + (((size_t)bb * NNN + n) * CC + c) * 16;
#pragma unroll
  for (int p = 0; p < 4; p++)
#pragma unroll
    for (int r = 0; r < 4; r++) {
      float acc = 0.f;
#pragma unroll
      for (int q = 0; q < 4; q++) acc = fmaf(wb[p * 4 + q], pn[q * 4 + r], acc);
      vout[p * 4 + r] = acc;
    }
}

// ---------------- stage 6: init R ----------------
__global__ void capsnet_fillR(float* __restrict__ R) {
  int gid = blockIdx.x * blockDim.x + threadIdx.x;
  if (gid < BATCH * NNN * CC) R[gid] = 0.1f;
}

// ---------------- stage 7: M-step (one wave32 per (b,c)) ----------------
// nkn layout: [0:1280)=Nj, [1280:2560)=kappa, [2560:3840)=nu
__global__ __launch_bounds__(32) void capsnet_mstep(const float* __restrict__ R,
                                                    const float* __restrict__ ai,
                                                    const float* __restrict__ V,
                                                    float* __restrict__ nkn,
                                                    float* __restrict__ m,
                                                    float* __restrict__ psi) {
  int bc = blockIdx.x;                 // 0..1279
  int bb = bc / CC, c = bc % CC;
  int lane = threadIdx.x;
  float sa = 0.f, sx[16], sxx[16];
#pragma unroll
  for (int d = 0; d < 16; d++) { sx[d] = 0.f; sxx[d] = 0.f; }
  for (int n = lane; n < NNN; n += 32) {
    float ra = R[((size_t)bb * NNN + n) * CC + c] * ai[(size_t)bb * NNN + n];
    const float* v = V + (((size_t)bb * NNN + n) * CC + c) * 16;
    sa += ra;
#pragma unroll
    for (int d = 0; d < 16; d++) { float vv = v[d]; sx[d] = fmaf(ra, vv, sx[d]); sxx[d] = fmaf(ra * vv, vv, sxx[d]); }
  }
#pragma unroll
  for (int o = 16; o > 0; o >>= 1) {
    sa += __shfl_xor(sa, o, 32);
#pragma unroll
    for (int d = 0; d < 16; d++) { sx[d] += __shfl_xor(sx[d], o, 32); sxx[d] += __shfl_xor(sxx[d], o, 32); }
  }
  float Nj  = sa + EPSF;
  float kap = 1.0f + Nj;               // KAPPA0 + Nj
  float nu  = 17.0f + Nj;              // NU0 + Nj, NU0 = D+1
  if (lane == 0) { nkn[bc] = Nj; nkn[1280 + bc] = kap; nkn[2560 + bc] = nu; }
  if (lane < 16) {
    int d = lane;
    float xb = sx[d] / Nj;
    float sg = (sxx[d] - 2.f * xb * sx[d] + xb * xb * sa) / Nj;
    m[bc * 16 + d]   = Nj * xb / kap;
    psi[bc * 16 + d] = 1.0f + Nj * sg + (Nj / kap) * xb * xb;   // KAPPA0 = 1
  }
}

// ---------------- stage 8: expectations (one wave32 per batch sample) -------------
__global__ __launch_bounds__(32) void capsnet_expect(const float* __restrict__ nkn,
                                                     const float* __restrict__ psi,
                                                     float* __restrict__ epi,
                                                     float* __restrict__ elam) {
  int bb = blockIdx.x;
  int lane = threadIdx.x;
  float alpha = 0.f, nu = 0.f;
  if (lane < CC) {
    float Nj = nkn[bb * CC + lane];
    alpha = 1.0f + Nj;                  // ALPHA0 + Nj
    nu = nkn[2560 + bb * CC + lane];
  }
  float asum = alpha;
#pragma unroll
  for (int o = 16; o > 0; o >>= 1) asum += __shfl_xor(asum, o, 32);
  if (lane < CC) {
    float e_pi = digammaf(alpha) - digammaf(asum);
    float acc = 0.f, lp = 0.f;
#pragma unroll
    for (int d = 1; d <= 16; d++) acc += digammaf(0.5f * (nu + 1.0f - (float)d));
#pragma unroll
    for (int d = 0; d < 16; d++) lp += logf(psi[(bb * CC + lane) * 16 + d]);
    epi[bb * CC + lane]  = e_pi;
    elam[bb * CC + lane] = acc + 16.0f * LN2 - lp;
  }
}

// ---------------- stage 9: E-step (softmax responsibilities) ----------------
__global__ void capsnet_estep(const float* __restrict__ V, const float* __restrict__ nkn,
                              const float* __restrict__ m, const float* __restrict__ psi,
                              const float* __restrict__ epi, const float* __restrict__ elam,
                              float* __restrict__ R) {
  int gid = blockIdx.x * blockDim.x + threadIdx.x;
  if (gid >= BATCH * NNN) return;
  int bb = gid / NNN;
  const float* vrow = V + (size_t)gid * CC * 16;
  float lnp[CC];
  float mx = -1e30f;
#pragma unroll
  for (int c = 0; c < CC; c++) {
    float kap = nkn[1280 + bb * CC + c];
    float nu  = nkn[2560 + bb * CC + c];
    const float* mc = m   + (bb * CC + c) * 16;
    const float* ps = psi + (bb * CC + c) * 16;
    float q = 0.f;
#pragma unroll
    for (int d = 0; d < 16; d++) { float dm = vrow[c * 16 + d] - mc[d]; q += dm * dm / ps[d]; }
    q = 16.0f / kap + nu * q;
    lnp[c] = epi[bb * CC + c] + 0.5f * elam[bb * CC + c] - 0.5f * 16.0f * LN2PI - 0.5f * q;
    mx = fmaxf(mx, lnp[c]);
  }
  float sum = 0.f;
#pragma unroll
  for (int c = 0; c < CC; c++) { lnp[c] = expf(lnp[c] - mx); sum += lnp[c]; }
  float inv = 1.f / sum;
#pragma unroll
  for (int c = 0; c < CC; c++) R[(size_t)gid * CC + c] = lnp[c] * inv;
}

// ---------------- stage 10: logits -> per-class batch-norm -> sigmoid -------------
__global__ __launch_bounds__(128) void capsnet_final(const float* __restrict__ epi,
                                                     const float* __restrict__ elam,
                                                     float* __restrict__ out) {
  int c = blockIdx.x;            // 0..9
  int bb = threadIdx.x;          // 0..127
  __shared__ float s1[128], s2[128];
  float logit = epi[bb * CC + c] + 0.5f * elam[bb * CC + c];
  s1[bb] = logit; s2[bb] = logit * logit;
  __syncthreads();
  for (int o = 64; o > 0; o >>= 1) {
    if (bb < o) { s1[bb] += s1[bb + o]; s2[bb] += s2[bb + o]; }
    __syncthreads();
  }
  float mu = s1[0] / 128.f, var = s2[0] / 128.f - mu * mu;
  float y = (logit - mu) * rsqrtf(var + BN_EPS);
  out[bb * CC + c] = 1.f / (1.f + expf(-y));
}

// ---------------- orchestration ----------------
extern "C" void kernel_launch(void* const* d_in, const int* in_sizes, int n_in,
                              void* d_out, int out_size, void* d_ws, size_t ws_size,
                              hipStream_t stream) {
  const float* x       = (const float*)d_in[0];
  const float* conv1_w = (const float*)d_in[1];
  const float* bn1_g   = (const float*)d_in[2];
  const float* bn1_b   = (const float*)d_in[3];
  const float* prim_w  = (const float*)d_in[4];
  const float* bna_g   = (const float*)d_in[5];
  const float* bna_b   = (const float*)d_in[6];
  const float* bnp_g   = (const float*)d_in[7];
  const float* bnp_b   = (const float*)d_in[8];
  const float* W_ij    = (const float*)d_in[9];
  float* out = (float*)d_out;
  float* ws = (float*)d_ws;

  // workspace layout (floats)
  size_t off = 0;
  float* h    = ws + off; off += (size_t)BATCH * ACH * HW1;       // 3,211,264
  float* col  = ws + off; off += (size_t)KGEMM * GN;              // 5,308,416
  float* pc   = ws + off; off += (size_t)BATCH * PCCH * SS;       // 2,506,752
  float* st1  = ws + off; off += 2 * ACH;                          // conv1 stats
  float* sta  = ws + off; off += 2 * BCAP;                         // activation stats
  float* stp  = ws + off; off += 2 * BCAP;                         // pose stats
  float* ai   = ws + off; off += (size_t)BATCH * NNN;              // 147,456
  float* V    = ws + off; off += (size_t)BATCH * NNN * CC * 16;    // 23,592,960
  float* R    = ws + off; off += (size_t)BATCH * NNN * CC;         // 1,474,560
  float* nkn  = ws + off; off += 3 * BATCH * CC;                   // Nj/kappa/nu
  float* mjs  = ws + off; off += (size_t)BATCH * CC * 16;
  float* psi  = ws + off; off += (size_t)BATCH * CC * 16;
  float* epi  = ws + off; off += BATCH * CC;
  float* elam = ws + off; off += BATCH * CC;
  (void)ws_size; (void)in_sizes; (void)n_in; (void)out_size;

  // 1) conv1
  {
    int n = BATCH * ACH * HW1;
    capsnet_conv1<<<(n + 255) / 256, 256, 0, stream>>>(x, conv1_w, h);
  }
  // 2) conv1 BN stats, then fused BN/ReLU + im2col materialization
  capsnet_stats_conv1<<<ACH, 256, 0, stream>>>(h, st1);
  {
    int n = KGEMM * GN;
    capsnet_bn_im2col<<<(n + 255) / 256, 256, 0, stream>>>(h, st1, bn1_g, bn1_b, col);
  }
  // 3) prim conv — WMMA GEMM on materialized operands, exact 16-tiles (34 x 288), K=1152
  capsnet_primconv_wmma<<<dim3(GN / 16, PCCH / 16), 32, 0, stream>>>(col, prim_w, pc);
  // 4) activation branch BN + sigmoid
  capsnet_stats_a<<<BCAP, 256, 0, stream>>>(pc, sta);
  {
    int n = BATCH * NNN;
    capsnet_apply_a<<<(n + 255) / 256, 256, 0, stream>>>(pc, sta, bna_g, bna_b, ai);
  }
  // 5) pose BN stats + vote transform
  capsnet_stats_p<<<BCAP, 256, 0, stream>>>(pc, stp);
  {
    int n = BATCH * BCAP * SS * CC;
    capsnet_votes<<<(n + 255) / 256, 256, 0, stream>>>(pc, stp, bnp_g, bnp_b, W_ij, V);
  }
  // 6) init responsibilities
  {
    int n = BATCH * NNN * CC;
    capsnet_fillR<<<(n + 255) / 256, 256, 0, stream>>>(R);
  }
  // 7) VB routing: 3 iterations
  for (int t = 0; t < 3; t++) {
    capsnet_mstep<<<BATCH * CC, 32, 0, stream>>>(R, ai, V, nkn, mjs, psi);
    if (t < 2) {
      capsnet_expect<<<BATCH, 32, 0, stream>>>(nkn, psi, epi, elam);
      int n = BATCH * NNN;
      capsnet_estep<<<(n + 255) / 256, 256, 0, stream>>>(V, nkn, mjs, psi, epi, elam, R);
    }
  }
  // 8) final expectations + logits BN + sigmoid
  capsnet_expect<<<BATCH, 32, 0, stream>>>(nkn, psi, epi, elam);
  capsnet_final<<<CC, 128, 0, stream>>>(epi, elam, out);
}